// DistSageConv_43860206027424
// MI455X (gfx1250) — compile-verified
//
#include <hip/hip_runtime.h>
#include <hip/hip_bf16.h>

// ---------------------------------------------------------------------------
// DistSAGEConv (mean aggregation) for MI455X / gfx1250, wave32.
//   out = feat @ W_self^T + (scatter_mean(feat[src] -> dst)) @ W_neigh^T + b
// Memory-bound: feat (25.6MB) + agg (25.6MB) are L2-resident (192MB L2); the
// GEMM is only 1.64 GFLOP -> keep full f32 and use V_WMMA_F32_16X16X4_F32.
// Weight staging into LDS uses the Tensor Data Mover (tensor_load_to_lds +
// s_wait_tensorcnt) with a manually-built D# descriptor.
// ---------------------------------------------------------------------------

typedef __attribute__((ext_vector_type(2))) float v2f;
typedef __attribute__((ext_vector_type(8))) float v8f;
typedef __attribute__((ext_vector_type(4))) unsigned int u32x4;
typedef __attribute__((ext_vector_type(4))) int i32x4;
typedef __attribute__((ext_vector_type(8))) int i32x8;

constexpr int D = 64; // D_IN == D_OUT == 64

#if defined(__has_builtin)
#if __has_builtin(__builtin_amdgcn_tensor_load_to_lds) && \
    __has_builtin(__builtin_amdgcn_s_wait_tensorcnt)
#define USE_TDM 1
#endif
#endif
#ifndef USE_TDM
#define USE_TDM 0
#endif

// ---------------- kernel 1: zero workspace (agg + deg contiguous) ----------
__global__ void zero_f32(float* __restrict__ p, size_t n) {
  size_t i = (size_t)blockIdx.x * blockDim.x + threadIdx.x;
  size_t stride = (size_t)gridDim.x * blockDim.x;
  for (; i < n; i += stride) p[i] = 0.0f;
}

// ---------------- kernel 2: transpose weights to k-major -------------------
// WT[k*64 + n] = W[n*64 + k]  (so WMMA B-fragment reads are n-contiguous).
// WTs and WTn are contiguous in the workspace -> one 32KB TDM tile later.
__global__ void transpose64(const float* __restrict__ Ws,
                            const float* __restrict__ Wn,
                            float* __restrict__ WTs,
                            float* __restrict__ WTn) {
  for (int i = threadIdx.x + blockIdx.x * blockDim.x; i < D * D;
       i += blockDim.x * gridDim.x) {
    int n = i >> 6;
    int k = i & 63;
    WTs[k * D + n] = Ws[n * D + k];
    WTn[k * D + n] = Wn[n * D + k];
  }
}

// ---------------- kernel 3: edge scatter-add (segment sum + degree) --------
// 16 threads per edge; each thread moves one float4 of the 64-float row.
// Gathers hit L2 (feat fits); atomics resolve in L2 atomic units.
__global__ void sage_scatter(const float* __restrict__ feat,
                             const int* __restrict__ src,
                             const int* __restrict__ dst,
                             float* __restrict__ agg,
                             float* __restrict__ deg,
                             int n_edges) {
  int gid = blockIdx.x * blockDim.x + threadIdx.x;
  int e = gid >> 4;
  int q = gid & 15;
  if (e >= n_edges) return;

  int s = src[e];
  int d = dst[e];

  // speculative prefetch of a later edge's source row (global_prefetch_b8)
  if (e + 8 < n_edges) {
    __builtin_prefetch(feat + (size_t)src[e + 8] * D, 0, 0);
  }

  const float4 v =
      ((const float4*)(feat + (size_t)s * D))[q]; // coalesced 16B/lane
  float* a = agg + (size_t)d * D + (size_t)q * 4;
  atomicAdd(a + 0, v.x); // global_atomic_add_f32 (no-return form)
  atomicAdd(a + 1, v.y);
  atomicAdd(a + 2, v.z);
  atomicAdd(a + 3, v.w);
  if (q == 0) atomicAdd(deg + d, 1.0f);
}

// ---- TDM helper: 1-row tensor copy  global -> LDS  (nelem f32, <=65535) ---
__device__ __forceinline__ void tdm_load_row_to_lds(const float* gptr,
                                                    unsigned lds_off,
                                                    unsigned nelem) {
#if USE_TDM
  unsigned long long ga = (unsigned long long)(uintptr_t)gptr;
  u32x4 g0;
  g0[0] = 1u;                                    // count=1 (valid user D#)
  g0[1] = lds_off;                               // lds_addr [63:32]
  g0[2] = (unsigned)(ga & 0xFFFFFFFFull);        // global_addr[31:0]
  g0[3] = (unsigned)((ga >> 32) & 0x1FFFFFFull)  // global_addr[56:32]
          | (2u << 30);                          // type=2 ("image")
  i32x8 g1;
  g1[0] = (int)(2u << 16);                       // data_size=2 -> 4 bytes
  g1[1] = (int)((nelem & 0xFFFFu) << 16);        // tensor_dim0[15:0] @ b48
  g1[2] = (int)(((nelem >> 16) & 0xFFFFu)        // tensor_dim0[31:16]
                | (1u << 16));                   // tensor_dim1 = 1
  g1[3] = (int)((nelem & 0xFFFFu) << 16);        // tile_dim0 = nelem
  g1[4] = 1;                                     // tile_dim1=1, tile_dim2=0
  g1[5] = (int)nelem;                            // tensor_dim0_stride[31:0]
  g1[6] = 0;
  g1[7] = 0;
  i32x4 z4 = {0, 0, 0, 0};
#if __clang_major__ >= 23
  i32x8 z8 = {0, 0, 0, 0, 0, 0, 0, 0};
  __builtin_amdgcn_tensor_load_to_lds(g0, g1, z4, z4, z8, 0);
#else
  __builtin_amdgcn_tensor_load_to_lds(g0, g1, z4, z4, 0);
#endif
#endif
}

// ---------------- kernel 4: fused mean + dual GEMM + bias via WMMA ---------
// One wave handles a 16-node tile: out[16,64].
// A (16x4 f32, ISA layout): lanes 0-15: M=lane, K={k0,k0+1}; lanes 16-31:
// M=lane-16, K={k0+2,k0+3}. B (4x16): lanes 0-15: N=lane, K rows {k0,k0+1};
// lanes 16-31: N=lane-16, K rows {k0+2,k0+3}. C/D: VGPR r -> M=r (lanes 0-15)
// and M=8+r (lanes 16-31), N = lane%16.
__global__ void __launch_bounds__(256)
sage_fused_gemm(const float* __restrict__ feat,
                const float* __restrict__ agg,
                const float* __restrict__ deg,
                const float* __restrict__ WT2, // k-major [W_self^T | W_neigh^T]
                const float* __restrict__ bias,
                float* __restrict__ out,
                int n_nodes) {
  __shared__ float ldsW[2 * D * D]; // 32 KB of the 320 KB WGP LDS
  const float* ldsWs = ldsW;
  const float* ldsWn = ldsW + D * D;

  const int lane = threadIdx.x & 31;
  const int wave = threadIdx.x >> 5;

#if USE_TDM
  if (wave == 0) {
    // One TDM descriptor copies both transposed weight matrices (32 KB).
    tdm_load_row_to_lds(WT2, (unsigned)(uintptr_t)(void*)ldsW, 2u * D * D);
    __builtin_amdgcn_s_wait_tensorcnt(0); // data in LDS before barrier
  }
#else
  for (int i = threadIdx.x; i < 2 * D * D; i += blockDim.x) {
    ldsW[i] = WT2[i];
  }
#endif
  __syncthreads();

  const int tile = blockIdx.x * (blockDim.x >> 5) + wave;
  const int row0 = tile * 16;
  if (row0 >= n_nodes) return; // uniform per wave: EXEC stays all-ones

  const int half = lane >> 4; // 0: K+{0,1}  1: K+{2,3}
  const int lid = lane & 15;  // M for A-frags, N for B/C/D-frags

  int mrow = row0 + lid;
  if (mrow >= n_nodes) mrow = n_nodes - 1; // clamp tail loads
  const float* featRow = feat + (size_t)mrow * D;
  const float* aggRow = agg + (size_t)mrow * D;
  const float invdeg = 1.0f / fmaxf(deg[mrow], 1.0f); // exact, as reference

  // Hoist all A-fragments for this tile (both operands), mean fused in.
  v2f aS[D / 4];
  v2f aN[D / 4];
#pragma unroll
  for (int ks = 0; ks < D / 4; ++ks) {
    const int ka = ks * 4 + half * 2;
    aS[ks] = v2f{featRow[ka], featRow[ka + 1]};
    aN[ks] = v2f{aggRow[ka] * invdeg, aggRow[ka + 1] * invdeg};
  }

#pragma unroll
  for (int nb = 0; nb < D; nb += 16) {
    const float bv = bias[nb + lid];
    v8f acc;
#pragma unroll
    for (int r = 0; r < 8; ++r) acc[r] = bv;

#pragma unroll
    for (int ks = 0; ks < D / 4; ++ks) {
      const int ka = ks * 4 + half * 2;
      v2f bS = v2f{ldsWs[ka * D + nb + lid], ldsWs[(ka + 1) * D + nb + lid]};
      v2f bN = v2f{ldsWn[ka * D + nb + lid], ldsWn[(ka + 1) * D + nb + lid]};
      // D = A*B + C   (f32, exact-precision path)
      acc = __builtin_amdgcn_wmma_f32_16x16x4_f32(
          false, aS[ks], false, bS, (short)0, acc, false, false);
      acc = __builtin_amdgcn_wmma_f32_16x16x4_f32(
          false, aN[ks], false, bN, (short)0, acc, false, false);
    }

#pragma unroll
    for (int r = 0; r < 8; ++r) {
      const int mm = row0 + r + half * 8;
      if (mm < n_nodes) out[(size_t)mm * D + nb + lid] = acc[r];
    }
  }
}

// ---------------------------------------------------------------------------
extern "C" void kernel_launch(void* const* d_in, const int* in_sizes, int n_in,
                              void* d_out, int out_size, void* d_ws,
                              size_t ws_size, hipStream_t stream) {
  const float* feat = (const float*)d_in[0];
  const int* src = (const int*)d_in[1];
  const int* dst = (const int*)d_in[2];
  const float* Ws = (const float*)d_in[3];
  const float* Wn = (const float*)d_in[4];
  const float* b = (const float*)d_in[5];
  float* out = (float*)d_out;

  const int n_nodes = in_sizes[0] / D;
  const int n_edges = in_sizes[1];

  // workspace layout: agg[N*64] | deg[N] | WT_self[64*64] WT_neigh[64*64]
  float* agg = (float*)d_ws;
  float* deg = agg + (size_t)n_nodes * D;
  float* WTs = deg + n_nodes;      // 16-byte aligned (26,000,000 % 16 == 0)
  float* WTn = WTs + D * D;        // contiguous with WTs -> single TDM tile

  const size_t zcount = (size_t)n_nodes * D + (size_t)n_nodes;
  zero_f32<<<1024, 256, 0, stream>>>(agg, zcount);
  transpose64<<<16, 256, 0, stream>>>(Ws, Wn, WTs, WTn);

  const long long scat_threads = (long long)n_edges * 16;
  const int scat_blocks = (int)((scat_threads + 255) / 256);
  sage_scatter<<<scat_blocks, 256, 0, stream>>>(feat, src, dst, agg, deg,
                                                n_edges);

  const int tiles = (n_nodes + 15) / 16; // 6250 tiles of 16 nodes
  const int blocks = (tiles + 7) / 8;    // 8 waves (tiles) per block
  sage_fused_gemm<<<blocks, 256, 0, stream>>>(feat, agg, deg, WTs, b, out,
                                              n_nodes);
}